// GMM_45646912422005
// MI455X (gfx1250) — compile-verified
//
#include <hip/hip_runtime.h>

typedef __attribute__((ext_vector_type(2))) float v2f;
typedef __attribute__((ext_vector_type(8))) float v8f;

#define NNODES 50000
#define NEDGES 800000

// ---------------------------------------------------------------------------
// GEMM: H[M, Nd] = X[M, Kd] @ W[Kd, Nd]   (fp32, V_WMMA_F32_16X16X4_F32)
// Block: 256 threads = 8 wave32; block tile 128(M) x 64(N); wave tile 16 x 64.
// Main loop is wave-uniform (no EXEC divergence) and uses strength-reduced
// marching pointers: A fragment = one aligned 8-byte load (+16B/step), B rows
// = two pointers advanced by 4*Nd floats/step (SGPR stride add, no v_mul_u64).
// Tail (only cin==3) uses lane-guarded scalar loads; EXEC reconverges before
// every WMMA.
// ---------------------------------------------------------------------------
__global__ __launch_bounds__(256)
void gemm_wmma_f32(const float* __restrict__ X, const float* __restrict__ W,
                   float* __restrict__ H, int M, int Kd, int Nd) {
  const int lane = threadIdx.x & 31;
  const int wave = threadIdx.x >> 5;
  const int l    = lane & 15;
  const int half = lane >> 4;                 // 0: K 0..1 / rows 0..7 ; 1: K 2..3 / rows 8..15
  const int m0   = blockIdx.x * 128 + wave * 16;
  const int n0   = blockIdx.y * 64;

  v8f acc0 = {}, acc1 = {}, acc2 = {}, acc3 = {};

  int arow = m0 + l; if (arow > M - 1) arow = M - 1;   // clamp (stores guarded)
  const float* __restrict__ xrow = X + (size_t)arow * Kd;
  const int col = n0 + l;

  // Marching pointers (strength-reduced addressing).
  const float* __restrict__ xr  = xrow + half * 2;               // +4 floats/step
  const float* __restrict__ wr0 = W + (size_t)(half * 2) * Nd + col;
  const float* __restrict__ wr1 = wr0 + Nd;
  const size_t wstep = (size_t)4 * (size_t)Nd;                   // floats/step

  int k = 0;
  // ---- main loop: wave-uniform trip count, no guards (Kd % 4 == 0 layers) ----
  for (; k + 4 <= Kd; k += 4) {
    const v2f a = *(const v2f*)xr;

    v2f b0, b1, b2, b3;
    b0.x = wr0[0];  b0.y = wr1[0];
    b1.x = wr0[16]; b1.y = wr1[16];
    b2.x = wr0[32]; b2.y = wr1[32];
    b3.x = wr0[48]; b3.y = wr1[48];

    acc0 = __builtin_amdgcn_wmma_f32_16x16x4_f32(false, a, false, b0, (short)0, acc0, false, false);
    acc1 = __builtin_amdgcn_wmma_f32_16x16x4_f32(false, a, false, b1, (short)0, acc1, false, false);
    acc2 = __builtin_amdgcn_wmma_f32_16x16x4_f32(false, a, false, b2, (short)0, acc2, false, false);
    acc3 = __builtin_amdgcn_wmma_f32_16x16x4_f32(false, a, false, b3, (short)0, acc3, false, false);

    xr  += 4;
    wr0 += wstep;
    wr1 += wstep;
  }

  // ---- tail (only taken when Kd % 4 != 0, i.e. cin == 3) ----
  if (k < Kd) {
    const int ka = k + half * 2;
    v2f a;
    a.x = (ka     < Kd) ? xrow[ka]     : 0.0f;
    a.y = (ka + 1 < Kd) ? xrow[ka + 1] : 0.0f;

    v2f b0 = {}, b1 = {}, b2 = {}, b3 = {};
    if (ka < Kd) {
      b0.x = wr0[0];  b1.x = wr0[16]; b2.x = wr0[32]; b3.x = wr0[48];
      if (ka + 1 < Kd) {
        b0.y = wr1[0];  b1.y = wr1[16]; b2.y = wr1[32]; b3.y = wr1[48];
      }
    }

    acc0 = __builtin_amdgcn_wmma_f32_16x16x4_f32(false, a, false, b0, (short)0, acc0, false, false);
    acc1 = __builtin_amdgcn_wmma_f32_16x16x4_f32(false, a, false, b1, (short)0, acc1, false, false);
    acc2 = __builtin_amdgcn_wmma_f32_16x16x4_f32(false, a, false, b2, (short)0, acc2, false, false);
    acc3 = __builtin_amdgcn_wmma_f32_16x16x4_f32(false, a, false, b3, (short)0, acc3, false, false);
  }

  const int rbase = m0 + half * 8;
  #pragma unroll
  for (int r = 0; r < 8; ++r) {
    const int row = rbase + r;
    if (row < M) {
      float* __restrict__ hrow = H + (size_t)row * Nd + col;
      hrow[0]  = acc0[r];
      hrow[16] = acc1[r];
      hrow[32] = acc2[r];
      hrow[48] = acc3[r];
    }
  }
}

// ---------------------------------------------------------------------------
// Per-edge Gaussian mixture weights: gw[e][k] = exp(-0.5*|(p - mu_k)*isig_k|^2)
// ---------------------------------------------------------------------------
__global__ __launch_bounds__(256)
void edge_gauss(const float* __restrict__ ea, const float* __restrict__ mu,
                const float* __restrict__ isig, float* __restrict__ gw, int E) {
  const int e = blockIdx.x * blockDim.x + threadIdx.x;
  if (e >= E) return;
  const float p0 = ea[3 * e + 0];
  const float p1 = ea[3 * e + 1];
  const float p2 = ea[3 * e + 2];
  float g[2];
  #pragma unroll
  for (int k = 0; k < 2; ++k) {
    const float d0 = (p0 - mu[3 * k + 0]) * isig[3 * k + 0];
    const float d1 = (p1 - mu[3 * k + 1]) * isig[3 * k + 1];
    const float d2 = (p2 - mu[3 * k + 2]) * isig[3 * k + 2];
    g[k] = __expf(-0.5f * (d0 * d0 + d1 * d1 + d2 * d2));
  }
  *(float2*)(gw + 2 * e) = make_float2(g[0], g[1]);
}

// ---------------------------------------------------------------------------
// In-degree (layer invariant): deg[dst[e]] += 1
// ---------------------------------------------------------------------------
__global__ __launch_bounds__(256)
void edge_deg(const int* __restrict__ dst, float* __restrict__ deg, int E) {
  const int e = blockIdx.x * blockDim.x + threadIdx.x;
  if (e < E) atomicAdd(&deg[dst[e]], 1.0f);
}

// ---------------------------------------------------------------------------
// Scatter with folded K-sum, 4 channels per thread:
//   agg[dst[e]][c..c+3] += H[src[e]][0][c..c+3]*gw[e][0]
//                        + H[src[e]][1][c..c+3]*gw[e][1]
// Two b128 gathers + 4 no-return f32 atomics per thread. H/agg stay
// L2-resident (<=153 MB < 192 MB L2) -> L2-bandwidth-bound.
// ---------------------------------------------------------------------------
__global__ __launch_bounds__(256)
void edge_scatter4(const float* __restrict__ H, const float* __restrict__ gw,
                   const int* __restrict__ src, const int* __restrict__ dst,
                   float* __restrict__ agg, int E, int cs) {
  const int idx   = blockIdx.x * blockDim.x + threadIdx.x;
  const int qs    = cs - 2;                   // log2(cout/4)
  const int total = E << qs;
  if (idx >= total) return;
  const int cout = 1 << cs;
  const int e = idx >> qs;
  const int c = (idx & ((1 << qs) - 1)) << 2;
  const int s = src[e];
  const int d = dst[e];
  const float2 g = *(const float2*)(gw + 2 * e);

  const float* __restrict__ hs = H + ((size_t)s << (cs + 1)) + c;
  const float4 h0 = *(const float4*)(hs);
  const float4 h1 = *(const float4*)(hs + cout);

  float* __restrict__ ag = agg + (((size_t)d << cs) + c);
  atomicAdd(ag + 0, h0.x * g.x + h1.x * g.y);
  atomicAdd(ag + 1, h0.y * g.x + h1.y * g.y);
  atomicAdd(ag + 2, h0.z * g.x + h1.z * g.y);
  atomicAdd(ag + 3, h0.w * g.x + h1.w * g.y);
}

// ---------------------------------------------------------------------------
// Finalize (4-wide): x_out[n][c..c+3] = agg[n][c..c+3] / max(deg[n],1) + b[c..c+3]
// ---------------------------------------------------------------------------
__global__ __launch_bounds__(256)
void node_finalize4(const float* __restrict__ agg, const float* __restrict__ deg,
                    const float* __restrict__ bias, float* __restrict__ xout,
                    int n_nodes, int cs) {
  const int idx   = blockIdx.x * blockDim.x + threadIdx.x;
  const int qs    = cs - 2;
  const int total = n_nodes << qs;
  if (idx >= total) return;
  const int n = idx >> qs;
  const int c = (idx & ((1 << qs) - 1)) << 2;

  const float inv = 1.0f / fmaxf(deg[n], 1.0f);
  const float4 a  = *(const float4*)(agg + ((size_t)idx << 2));
  const float4 b  = *(const float4*)(bias + c);
  float4 o;
  o.x = a.x * inv + b.x;
  o.y = a.y * inv + b.y;
  o.z = a.z * inv + b.z;
  o.w = a.w * inv + b.w;
  *(float4*)(xout + ((size_t)idx << 2)) = o;
}

// ---------------------------------------------------------------------------
// Driver
// ---------------------------------------------------------------------------
extern "C" void kernel_launch(void* const* d_in, const int* in_sizes, int n_in,
                              void* d_out, int out_size, void* d_ws, size_t ws_size,
                              hipStream_t stream) {
  (void)in_sizes; (void)n_in; (void)out_size; (void)ws_size;

  const float* feat = (const float*)d_in[0];
  const float* ea   = (const float*)d_in[1];
  const int*   src  = (const int*)d_in[2];
  const int*   dst  = (const int*)d_in[3];
  const float* mu[4]  = {(const float*)d_in[4],  (const float*)d_in[8],
                         (const float*)d_in[12], (const float*)d_in[16]};
  const float* isg[4] = {(const float*)d_in[5],  (const float*)d_in[9],
                         (const float*)d_in[13], (const float*)d_in[17]};
  const float* Wm[4]  = {(const float*)d_in[6],  (const float*)d_in[10],
                         (const float*)d_in[14], (const float*)d_in[18]};
  const float* bv[4]  = {(const float*)d_in[7],  (const float*)d_in[11],
                         (const float*)d_in[15], (const float*)d_in[19]};

  char* ws = (char*)d_ws;
  float* H   = (float*)ws;                                   // N*512 f32 (102.4 MB)
  float* agg = (float*)(ws + (size_t)NNODES * 512 * 4);      // N*256 f32 ( 51.2 MB)
  float* gw  = (float*)(ws + (size_t)NNODES * 768 * 4);      // E*2   f32 (  6.4 MB)
  float* deg = gw + (size_t)NEDGES * 2;                      // N     f32
  float* xA  = deg + NNODES;                                 // N*256 f32 ( 51.2 MB)

  const int cin[4] = {3, 64, 128, 256};
  const int cs [4] = {6, 7, 8, 8};                           // log2(cout)

  const float* xin [4] = {feat, xA, (const float*)d_out, xA};
  float*       xout[4] = {xA, (float*)d_out, xA, (float*)d_out};

  // degree (shared by all layers)
  hipMemsetAsync(deg, 0, NNODES * sizeof(float), stream);
  edge_deg<<<(NEDGES + 255) / 256, 256, 0, stream>>>(dst, deg, NEDGES);

  for (int lyr = 0; lyr < 4; ++lyr) {
    const int cout = 1 << cs[lyr];
    const int Nd   = 2 * cout;

    hipMemsetAsync(agg, 0, (size_t)NNODES * cout * sizeof(float), stream);

    edge_gauss<<<(NEDGES + 255) / 256, 256, 0, stream>>>(ea, mu[lyr], isg[lyr], gw, NEDGES);

    dim3 grid((NNODES + 127) / 128, Nd / 64);
    gemm_wmma_f32<<<grid, 256, 0, stream>>>(xin[lyr], Wm[lyr], H, NNODES, cin[lyr], Nd);

    const int etotal = NEDGES << (cs[lyr] - 2);
    edge_scatter4<<<(etotal + 255) / 256, 256, 0, stream>>>(H, gw, src, dst, agg,
                                                            NEDGES, cs[lyr]);

    const int ftotal = NNODES << (cs[lyr] - 2);
    node_finalize4<<<(ftotal + 255) / 256, 256, 0, stream>>>(agg, deg, bv[lyr],
                                                             xout[lyr], NNODES, cs[lyr]);
  }
}